// AttnPooling_3143916061233
// MI455X (gfx1250) — compile-verified
//
#include <hip/hip_runtime.h>
#include <hip/hip_bf16.h>

// Problem constants (B=8, N=16, D=128, H=W=64)
#define BN 128          // B*N pairs
#define DC 128          // channels
#define KC 4096         // tokens per pair (H*W)

typedef __attribute__((ext_vector_type(2))) float v2f;
typedef __attribute__((ext_vector_type(8))) float v8f;

__device__ __forceinline__ float wave_sum(float v) {
#pragma unroll
  for (int o = 16; o > 0; o >>= 1) v += __shfl_xor(v, o, 32);
  return v;
}
__device__ __forceinline__ float wave_max(float v) {
#pragma unroll
  for (int o = 16; o > 0; o >>= 1) v = fmaxf(v, __shfl_xor(v, o, 32));
  return v;
}

// ---------------------------------------------------------------------------
// Pass 1: masked mean over tokens.  grid=(BN, DC/16), block=256 (8 waves).
// ---------------------------------------------------------------------------
__global__ __launch_bounds__(256)
void mean_kernel(const float* __restrict__ x, const int* __restrict__ mask,
                 float* __restrict__ meanOut) {
  __shared__ float sM[KC];      // 16 KB mask-as-float
  __shared__ float sRed[8];
  __shared__ float sCount;
  const int bn  = blockIdx.x;
  const int tid = threadIdx.x, lane = tid & 31, wave = tid >> 5;
  const float* xb = x + (size_t)bn * DC * KC;
  const int*   mb = mask + (size_t)bn * KC;

  float cnt = 0.f;
#pragma unroll
  for (int i = 0; i < 4; ++i) {
    const int k = tid * 4 + i * 1024;
    const int4 mv = *reinterpret_cast<const int4*>(mb + k);
    const float4 mf = make_float4((float)mv.x, (float)mv.y, (float)mv.z, (float)mv.w);
    *reinterpret_cast<float4*>(&sM[k]) = mf;
    cnt += mf.x + mf.y + mf.z + mf.w;
  }
  cnt = wave_sum(cnt);
  if (lane == 0) sRed[wave] = cnt;
  __syncthreads();
  if (tid == 0) {
    float r = 0.f;
#pragma unroll
    for (int i = 0; i < 8; ++i) r += sRed[i];
    sCount = r;
  }
  __syncthreads();
  const float count = sCount;

#pragma unroll
  for (int i = 0; i < 2; ++i) {
    const int e = blockIdx.y * 16 + wave * 2 + i;
    const float* xr = xb + (size_t)e * KC;
    float acc = 0.f;
    for (int k0 = lane * 4; k0 < KC; k0 += 128) {
      const float4 xv = *reinterpret_cast<const float4*>(xr + k0);
      const float4 wv = *reinterpret_cast<const float4*>(&sM[k0]);
      acc = fmaf(xv.x, wv.x, acc); acc = fmaf(xv.y, wv.y, acc);
      acc = fmaf(xv.z, wv.z, acc); acc = fmaf(xv.w, wv.w, acc);
    }
    acc = wave_sum(acc);
    if (lane == 0) meanOut[bn * DC + e] = acc / count;
  }
}

// ---------------------------------------------------------------------------
// Projection GEMM via V_WMMA_F32_16X16X4_F32 (fp32 tensor path).
// Out[M=128, N=128] = A[128,128] @ B + bias, where
//   TRANSB=1: B[k][n] = Bm[n*128 + k]   (y = A @ Bm^T, torch Linear)
//   TRANSB=0: B[k][n] = Bm[k*128 + n]
// grid=(8,8) tiles, one wave (32 threads) per 16x16 tile, 32 k-steps of 4.
// TRANSB is a compile-time template arg so the unrolled k-loop is pure
// straight-line load->wmma (no per-step scalar branching).
// ---------------------------------------------------------------------------
template <int TRANSB, int HAS_BIAS>
__global__ __launch_bounds__(32)
void proj_wmma(const float* __restrict__ A, const float* __restrict__ Bm,
               const float* __restrict__ bias, float* __restrict__ Out) {
  const int lane = threadIdx.x;
  const int half = lane >> 4;
  const int l15  = lane & 15;
  const int row  = blockIdx.x * 16 + l15;
  const int col  = blockIdx.y * 16 + l15;

  v8f acc = {0.f, 0.f, 0.f, 0.f, 0.f, 0.f, 0.f, 0.f};
#pragma unroll
  for (int k0 = 0; k0 < DC; k0 += 4) {
    const int ka = k0 + half * 2;
    // A fragment: contiguous pair -> one b64 load
    const v2f a = *reinterpret_cast<const v2f*>(A + row * DC + ka);
    v2f b;
    if (TRANSB) {
      // contiguous pair along k -> one b64 load
      b = *reinterpret_cast<const v2f*>(Bm + col * DC + ka);
    } else {
      b[0] = Bm[ka * DC + col];
      b[1] = Bm[(ka + 1) * DC + col];
    }
    acc = __builtin_amdgcn_wmma_f32_16x16x4_f32(false, a, false, b,
                                                (short)0, acc, false, false);
  }
  const float bv = HAS_BIAS ? bias[col] : 0.f;
#pragma unroll
  for (int r = 0; r < 8; ++r) {
    Out[(blockIdx.x * 16 + half * 8 + r) * DC + col] = acc[r] + bv;
  }
}

// c[bn] = query[bn] . bk   (tiny)
__global__ void cdot_kernel(const float* __restrict__ Q,
                            const float* __restrict__ bk,
                            float* __restrict__ cArr) {
  const int bn = threadIdx.x;
  float s = 0.f;
  for (int d = 0; d < DC; ++d) s = fmaf(Q[bn * DC + d], bk[d], s);
  cArr[bn] = s;
}

// ---------------------------------------------------------------------------
// Pass 2: logits[k] = (c + sum_e q'[e]*x[e][k]) * (1/sqrt(D)), masked -> -inf.
// grid=(BN, KC/1024), block=256, 4 tokens per thread (float4 coalesced).
// ---------------------------------------------------------------------------
__global__ __launch_bounds__(256)
void logits_kernel(const float* __restrict__ x, const int* __restrict__ mask,
                   const float* __restrict__ Qp, const float* __restrict__ cArr,
                   float* __restrict__ logits) {
  __shared__ float sQp[DC];
  const int bn  = blockIdx.x;
  const int tid = threadIdx.x;
  if (tid < DC) sQp[tid] = Qp[bn * DC + tid];
  __syncthreads();
  const float* xb = x + (size_t)bn * DC * KC;
  const int kbase = blockIdx.y * 1024 + tid * 4;
  const float cb  = cArr[bn];
  float l0 = cb, l1 = cb, l2 = cb, l3 = cb;
#pragma unroll 4
  for (int e = 0; e < DC; ++e) {
    const float q = sQp[e];
    const float4 xv = *reinterpret_cast<const float4*>(xb + (size_t)e * KC + kbase);
    l0 = fmaf(q, xv.x, l0); l1 = fmaf(q, xv.y, l1);
    l2 = fmaf(q, xv.z, l2); l3 = fmaf(q, xv.w, l3);
  }
  const float scale = 0.08838834764831845f;  // 1/sqrt(128)
  const int4 mv = *reinterpret_cast<const int4*>(mask + (size_t)bn * KC + kbase);
  float4 ov;
  ov.x = mv.x ? l0 * scale : -__builtin_inff();
  ov.y = mv.y ? l1 * scale : -__builtin_inff();
  ov.z = mv.z ? l2 * scale : -__builtin_inff();
  ov.w = mv.w ? l3 * scale : -__builtin_inff();
  *reinterpret_cast<float4*>(logits + (size_t)bn * KC + kbase) = ov;
}

// ---------------------------------------------------------------------------
// Pass 3: softmax over 4096 logits per pair (in-place -> normalized weights).
// ---------------------------------------------------------------------------
__global__ __launch_bounds__(1024)
void softmax_kernel(float* __restrict__ logits) {
  __shared__ float sRed[32];
  __shared__ float sB;
  const int bn  = blockIdx.x;
  const int tid = threadIdx.x, lane = tid & 31, wave = tid >> 5;
  float* lb = logits + (size_t)bn * KC;
  const float4 lv = *reinterpret_cast<const float4*>(lb + tid * 4);

  float mx = fmaxf(fmaxf(lv.x, lv.y), fmaxf(lv.z, lv.w));
  mx = wave_max(mx);
  if (lane == 0) sRed[wave] = mx;
  __syncthreads();
  if (wave == 0) {
    float r = sRed[lane];
    r = wave_max(r);
    if (lane == 0) sB = r;
  }
  __syncthreads();
  const float M = sB;

  const float w0 = __expf(lv.x - M), w1 = __expf(lv.y - M);
  const float w2 = __expf(lv.z - M), w3 = __expf(lv.w - M);
  float s = w0 + w1 + w2 + w3;
  s = wave_sum(s);
  if (lane == 0) sRed[wave] = s;
  __syncthreads();
  if (wave == 0) {
    float r = sRed[lane];
    r = wave_sum(r);
    if (lane == 0) sB = r;
  }
  __syncthreads();
  const float inv = 1.0f / sB;
  *reinterpret_cast<float4*>(lb + tid * 4) =
      make_float4(w0 * inv, w1 * inv, w2 * inv, w3 * inv);
}

// ---------------------------------------------------------------------------
// Pass 4: pooled[e] = sum_k w[k] * x[e][k].  grid=(BN, DC/16), block=256.
// ---------------------------------------------------------------------------
__global__ __launch_bounds__(256)
void pooled_kernel(const float* __restrict__ x, const float* __restrict__ wgt,
                   float* __restrict__ out) {
  const int bn  = blockIdx.x;
  const int tid = threadIdx.x, lane = tid & 31, wave = tid >> 5;
  const float* xb = x + (size_t)bn * DC * KC;
  const float* wb = wgt + (size_t)bn * KC;
#pragma unroll
  for (int i = 0; i < 2; ++i) {
    const int e = blockIdx.y * 16 + wave * 2 + i;
    const float* xr = xb + (size_t)e * KC;
    float acc = 0.f;
    for (int k0 = lane * 4; k0 < KC; k0 += 128) {
      const float4 xv = *reinterpret_cast<const float4*>(xr + k0);
      const float4 wv = *reinterpret_cast<const float4*>(wb + k0);
      acc = fmaf(xv.x, wv.x, acc); acc = fmaf(xv.y, wv.y, acc);
      acc = fmaf(xv.z, wv.z, acc); acc = fmaf(xv.w, wv.w, acc);
    }
    acc = wave_sum(acc);
    if (lane == 0) out[bn * DC + e] = acc;
  }
}

extern "C" void kernel_launch(void* const* d_in, const int* in_sizes, int n_in,
                              void* d_out, int out_size, void* d_ws, size_t ws_size,
                              hipStream_t stream) {
  const float* x    = (const float*)d_in[0];
  const int*   mask = (const int*)d_in[1];
  const float* Wq   = (const float*)d_in[2];
  const float* bq   = (const float*)d_in[3];
  const float* Wk   = (const float*)d_in[4];
  const float* bk   = (const float*)d_in[5];
  float* out = (float*)d_out;

  float* ws     = (float*)d_ws;
  float* meanB  = ws;                 // 128*128
  float* Qbuf   = ws + 16384;         // 128*128
  float* Qp     = ws + 32768;         // 128*128
  float* cArr   = ws + 49152;         // 128
  float* logits = ws + 49280;         // 128*4096 (becomes weights in-place)

  // 1) masked mean per (b,n)
  mean_kernel<<<dim3(BN, DC / 16), 256, 0, stream>>>(x, mask, meanB);
  // 2) Q = Mean @ Wq^T + bq   (fp32 WMMA)
  proj_wmma<1, 1><<<dim3(8, 8), 32, 0, stream>>>(meanB, Wq, bq, Qbuf);
  // 3) q' = Q @ Wk            (fp32 WMMA)  — folds keys-matmul into the query
  proj_wmma<0, 0><<<dim3(8, 8), 32, 0, stream>>>(Qbuf, Wk, nullptr, Qp);
  // 4) c = Q . bk
  cdot_kernel<<<1, DC, 0, stream>>>(Qbuf, bk, cArr);
  // 5) masked, scaled logits
  logits_kernel<<<dim3(BN, KC / 1024), 256, 0, stream>>>(x, mask, Qp, cArr, logits);
  // 6) softmax -> normalized weights (in place)
  softmax_kernel<<<BN, 1024, 0, stream>>>(logits);
  // 7) weighted pooling
  pooled_kernel<<<dim3(BN, DC / 16), 256, 0, stream>>>(x, logits, out);
}